// LAEconv_32298154066790
// MI455X (gfx1250) — compile-verified
//
#include <hip/hip_runtime.h>
#include <hip/hip_bf16.h>
#include <math.h>

// Problem constants (from reference): B=2, C=64, N=16384, K=32, O=64
#define Bsz  2
#define Cdim 64
#define Npts 16384
#define Kn   32
#define Odim 64
#define PITCH 17   // LDS row pitch (coprime with 64 banks -> conflict-free)

typedef float v2f __attribute__((ext_vector_type(2)));
typedef float v8f __attribute__((ext_vector_type(8)));

// ---------------------------------------------------------------------------
// prep: v[c] = sum_o alpha[o] * W[o,c]   (Wᵀ alpha, 64)
//       M[p,c] = sum_o mlp[p,o] * W[o,c] (mlp_w @ W_w, 64x64)
// ---------------------------------------------------------------------------
__global__ __launch_bounds__(256) void lae_prep(const float* __restrict__ W,
                                                const float* __restrict__ alpha,
                                                const float* __restrict__ mlp,
                                                float* __restrict__ v,
                                                float* __restrict__ M) {
  int t = threadIdx.x;
  if (t < Cdim) {
    float s = 0.f;
    for (int o = 0; o < Odim; ++o) s = fmaf(alpha[o], W[o * Cdim + t], s);
    v[t] = s;
  }
  for (int e = t; e < Odim * Cdim; e += 256) {
    int p = e >> 6, c = e & 63;
    float s = 0.f;
    for (int o = 0; o < Odim; ++o) s = fmaf(mlp[p * Odim + o], W[o * Cdim + c], s);
    M[e] = s;
  }
}

// ---------------------------------------------------------------------------
// transpose fea [B,C,N] -> feaT [B,N,C]  (LDS tiled, coalesced both sides)
// ---------------------------------------------------------------------------
__global__ __launch_bounds__(256) void lae_transpose(const float* __restrict__ fea,
                                                     float* __restrict__ feaT) {
  __shared__ float tile[64][65];
  int blk = blockIdx.x;
  int b  = blk >> 8;              // Npts/64 = 256 tiles per batch
  int n0 = (blk & 255) * 64;
  int tn = threadIdx.x & 63;
  int tr = threadIdx.x >> 6;      // 0..3
  for (int c = tr; c < 64; c += 4)
    tile[c][tn] = fea[((size_t)b * Cdim + c) * Npts + n0 + tn];
  __syncthreads();
  for (int nr = tr; nr < 64; nr += 4)
    feaT[((size_t)b * Npts + n0 + nr) * Cdim + tn] = tile[tn][nr];
}

// ---------------------------------------------------------------------------
// main: per wave (32 lanes) handle 16 points.
//  Phase 1 (per point): lane k = neighbor k -> gather vector, logit = v.g_k,
//    wave32 softmax via shfl_xor; then lane c = channel -> agg[c] = sum attn*g,
//    written into per-wave LDS tile [64 x 16] (pitch 17).
//  Phase 2: out(64x16) = M(64x64) x agg(64x16) via V_WMMA_F32_16X16X4_F32,
//    bias + ReLU fused on store.
// ---------------------------------------------------------------------------
__global__ __launch_bounds__(256) void lae_main(const float* __restrict__ feaT,
                                                const int* __restrict__ idx,
                                                const float* __restrict__ v,
                                                const float* __restrict__ M,
                                                const float* __restrict__ bias,
                                                float* __restrict__ out) {
  __shared__ float aggL[8][64 * PITCH];     // 34.8 KB
  const int lane  = threadIdx.x & 31;
  const int wave  = threadIdx.x >> 5;
  const int group = blockIdx.x * 8 + wave;  // 16-point group
  const int bn0   = group * 16;             // flat b*N + n base
  const int b     = bn0 >> 14;              // / Npts
  const int n0    = bn0 & (Npts - 1);
  float* agw = &aggL[wave][0];
  const float* gbase = feaT + (size_t)b * Npts * Cdim;

  for (int pt = 0; pt < 16; ++pt) {
    const int bn = bn0 + pt;
    int j = idx[(size_t)bn * Kn + lane];            // coalesced 32 ints
    const float* g = gbase + (size_t)j * Cdim;      // 256B contiguous vector
    float dot = 0.f;
#pragma unroll
    for (int c4 = 0; c4 < Cdim; c4 += 4) {
      float4 gv = *(const float4*)(g + c4);
      float4 vv = *(const float4*)(v + c4);         // wave-uniform -> scalar
      dot = fmaf(gv.x, vv.x, dot);
      dot = fmaf(gv.y, vv.y, dot);
      dot = fmaf(gv.z, vv.z, dot);
      dot = fmaf(gv.w, vv.w, dot);
    }
    // softmax over 32 neighbors (center term cancels: shift invariance)
    float mx = dot;
#pragma unroll
    for (int s = 16; s > 0; s >>= 1) mx = fmaxf(mx, __shfl_xor(mx, s, 32));
    float e = __expf(dot - mx);
    float sum = e;
#pragma unroll
    for (int s = 16; s > 0; s >>= 1) sum += __shfl_xor(sum, s, 32);
    float attn = e / sum;

    // lane = channel; broadcast (attn_k, j_k) from lane k
    float a0 = 0.f, a1 = 0.f;
#pragma unroll 4
    for (int k = 0; k < Kn; ++k) {
      float ak = __shfl(attn, k, 32);
      int   jk = __shfl(j, k, 32);
      const float* gk = gbase + (size_t)jk * Cdim;
      a0 = fmaf(ak, gk[lane], a0);                  // coalesced 128B
      a1 = fmaf(ak, gk[lane + 32], a1);
    }
    agw[lane * PITCH + pt]        = a0;             // pitch-17: no bank conflict
    agw[(lane + 32) * PITCH + pt] = a1;
  }
  __syncthreads();

  // ---- WMMA f32 16x16x4: D(16x16) += A(16x4: rows of M) x B(4x16: agg) ----
  const int half = lane >> 4;      // 0: K=0..1, 1: K=2..3 (per ISA A layout)
  const int l4   = lane & 15;
  const int koff = half * 2;
#pragma unroll
  for (int ob = 0; ob < 4; ++ob) {
    const int arow = ob * 16 + l4;                  // lanes 0-15 & 16-31: M=l4
    const float* Mr = M + arow * Cdim + koff;
    v8f acc = {0.f, 0.f, 0.f, 0.f, 0.f, 0.f, 0.f, 0.f};
#pragma unroll
    for (int c4 = 0; c4 < Cdim; c4 += 4) {
      float2 am = *(const float2*)(Mr + c4);
      v2f a;  a.x = am.x;  a.y = am.y;
      v2f bf;
      bf.x = agw[(c4 + koff)     * PITCH + l4];     // B row K=koff,   col=l4
      bf.y = agw[(c4 + koff + 1) * PITCH + l4];     // B row K=koff+1
      acc = __builtin_amdgcn_wmma_f32_16x16x4_f32(false, a, false, bf,
                                                  (short)0, acc, false, false);
    }
    // D layout: VGPR r -> (M=r, N=lane) lanes 0-15; (M=r+8) lanes 16-31
#pragma unroll
    for (int r = 0; r < 8; ++r) {
      int o = ob * 16 + half * 8 + r;
      float val = acc[r] + bias[o];
      out[((size_t)b * Odim + o) * Npts + n0 + l4] = fmaxf(val, 0.f);
    }
  }
}

// ---------------------------------------------------------------------------
extern "C" void kernel_launch(void* const* d_in, const int* in_sizes, int n_in,
                              void* d_out, int out_size, void* d_ws, size_t ws_size,
                              hipStream_t stream) {
  const float* fea     = (const float*)d_in[0];
  const int*   idx     = (const int*)  d_in[1];
  const float* W_w     = (const float*)d_in[2];
  const float* alpha_w = (const float*)d_in[3];
  const float* mlp_w   = (const float*)d_in[4];
  const float* mlp_b   = (const float*)d_in[5];
  float* out = (float*)d_out;

  // workspace layout: v[64] | M[64*64] | feaT[B*N*C]  (~8.4 MB)
  float* v    = (float*)d_ws;
  float* M    = v + 64;
  float* feaT = M + Odim * Cdim;

  lae_prep<<<1, 256, 0, stream>>>(W_w, alpha_w, mlp_w, v, M);
  lae_transpose<<<Bsz * (Npts / 64), 256, 0, stream>>>(fea, feaT);
  lae_main<<<(Bsz * Npts) / 128, 256, 0, stream>>>(feaT, idx, v, M, mlp_b, out);

  (void)in_sizes; (void)n_in; (void)out_size; (void)ws_size;
}